// PagedAttention_71786083385649
// MI455X (gfx1250) — compile-verified
//
#include <hip/hip_runtime.h>
#include <math.h>

// Paged-attention decode for MI455X (gfx1250, wave32).
// HBM-bound (AI ~2 FLOP/byte, ~512MB gathered KV): flash-decoding with
// 4-way context split, fp32 WMMA 16x16x4 for QK^T and P*V.

#define B_SZ    16
#define H_SZ    32
#define KVH_SZ  8
#define G_SZ    4
#define D_SZ    128
#define MAXCTX  4096
#define NW      8
#define THREADS (NW * 32)
#define PARTSZ  (8 + G_SZ * D_SZ)   // per (bk,split): m[4], l[4], acc[4][128]
#define SCALE   0.08838834764831845f
#define NEG_BIG (-3.0e38f)

typedef __attribute__((ext_vector_type(2))) float v2f;
typedef __attribute__((ext_vector_type(8))) float v8f;

__global__ __launch_bounds__(THREADS) void pa_main_kernel(
    const float* __restrict__ q,
    const float* __restrict__ kc,
    const float* __restrict__ vc,
    const int*   __restrict__ slots,
    const int*   __restrict__ ctx_lens,
    float*       __restrict__ out,
    float*       __restrict__ partial,
    int nsplits)
{
  __shared__ float Qs[16][132];          // zero-padded Q (B-matrix), stride 132 -> conflict-free
  __shared__ float Ps[NW][16][20];       // per-wave P tile [head][token], stride 20 -> conflict-free
  __shared__ float Rm[NW][G_SZ];
  __shared__ float Rl[NW][G_SZ];
  __shared__ float Racc[NW][G_SZ][D_SZ];

  const int wg    = blockIdx.x;
  const int split = wg % nsplits;
  const int bk    = wg / nsplits;
  const int b     = bk / KVH_SZ;
  const int kvh   = bk % KVH_SZ;
  const int tid   = threadIdx.x;
  const int w     = tid >> 5;
  const int lane  = tid & 31;
  const int lo    = lane & 15;
  const int hi    = lane >> 4;

  const int ctx    = ctx_lens[b];
  const int ntiles = (ctx + 15) >> 4;

  // Stage Q into LDS; rows >= G are zero (pad B-matrix columns 4..15).
  for (int i = tid; i < 16 * 132; i += THREADS) {
    int h = i / 132, dd = i - h * 132;
    float val = 0.f;
    if (h < G_SZ && dd < D_SZ)
      val = q[((size_t)b * H_SZ + kvh * G_SZ + h) * D_SZ + dd];
    Qs[h][dd] = val;
  }
  __syncthreads();

  const v8f vzero = {0.f, 0.f, 0.f, 0.f, 0.f, 0.f, 0.f, 0.f};
  float m_run = NEG_BIG;
  float l_run = 0.f;
  v8f acc[8];                            // O^T accumulators: d = 16*dblk + v + 8*hi, head = lo
#pragma unroll
  for (int i = 0; i < 8; i++) acc[i] = vzero;

  const int* srow = slots + (size_t)b * MAXCTX;

  for (int tile = split * NW + w; tile < ntiles; tile += nsplits * NW) {
    const int tbase = tile << 4;
    const int slot  = srow[tbase + lo];                       // this lane's token row
    const float* krow = kc + ((size_t)slot * KVH_SZ + kvh) * D_SZ;

    // Prefetch next tile's K/V rows (uniform branch; emits global_prefetch_b8).
    int ntile = tile + nsplits * NW;
    if (ntile < ntiles) {
      int nslot = srow[(ntile << 4) + lo];
      __builtin_prefetch(kc + ((size_t)nslot * KVH_SZ + kvh) * D_SZ, 0, 0);
      __builtin_prefetch(vc + ((size_t)nslot * KVH_SZ + kvh) * D_SZ, 0, 0);
    }

    // ---- QK^T: S^T[token][head] via 32 WMMA f32 16x16x4 steps over D=128 ----
    v8f s0 = vzero, s1 = vzero;
#pragma unroll
    for (int kk = 0; kk < 32; kk += 2) {
      v2f a0 = *(const v2f*)(krow + (kk * 4 + 2 * hi));               // A: K rows (gather b64)
      v2f b0 = *(const v2f*)(&Qs[lo][kk * 4 + 2 * hi]);               // B: Q from LDS
      s0 = __builtin_amdgcn_wmma_f32_16x16x4_f32(false, a0, false, b0, (short)0, s0, false, false);
      v2f a1 = *(const v2f*)(krow + ((kk + 1) * 4 + 2 * hi));
      v2f b1 = *(const v2f*)(&Qs[lo][(kk + 1) * 4 + 2 * hi]);
      s1 = __builtin_amdgcn_wmma_f32_16x16x4_f32(false, a1, false, b1, (short)0, s1, false, false);
    }

    // lane holds S^T[token = j + 8*hi][head = lo] for j=0..7
    float sc[8];
    float tmax = NEG_BIG;
#pragma unroll
    for (int j = 0; j < 8; j++) {
      float vsc = (s0[j] + s1[j]) * SCALE;
      int stok = tbase + j + 8 * hi;
      vsc = (stok < ctx) ? vsc : NEG_BIG;                             // mask, branch-free
      sc[j] = vsc;
      tmax = fmaxf(tmax, vsc);
    }
    tmax = fmaxf(tmax, __shfl_xor(tmax, 16, 32));                     // combine token halves
    float m_new = fmaxf(m_run, tmax);
    float f = __expf(m_run - m_new);
    float psum = 0.f;
#pragma unroll
    for (int j = 0; j < 8; j++) {
      float p = __expf(sc[j] - m_new);
      sc[j] = p;
      psum += p;
    }
    psum += __shfl_xor(psum, 16, 32);
    l_run = l_run * f + psum;
    m_run = m_new;
#pragma unroll
    for (int i = 0; i < 8; i++)
#pragma unroll
      for (int j = 0; j < 8; j++) acc[i][j] *= f;

    // Stage P into per-wave LDS as [head][token] (token pairs contiguous).
#pragma unroll
    for (int j = 0; j < 8; j += 2) {
      v2f pp; pp.x = sc[j]; pp.y = sc[j + 1];
      *(v2f*)(&Ps[w][lo][j + 8 * hi]) = pp;
    }

    // B-frags for PV: b = {P[4t+2hi][lo], P[4t+2hi+1][lo]} -> ds_load_b64
    v2f pb[4];
#pragma unroll
    for (int t = 0; t < 4; t++)
      pb[t] = *(const v2f*)(&Ps[w][lo][4 * t + 2 * hi]);

    // V row bases for the two token rows each A-frag needs (broadcast via shfl of slot).
    const float* vr0[4];
    const float* vr1[4];
#pragma unroll
    for (int t = 0; t < 4; t++) {
      int t0  = 4 * t + 2 * hi;
      int sl0 = __shfl(slot, t0, 32);
      int sl1 = __shfl(slot, t0 + 1, 32);
      vr0[t] = vc + ((size_t)sl0 * KVH_SZ + kvh) * D_SZ;
      vr1[t] = vc + ((size_t)sl1 * KVH_SZ + kvh) * D_SZ;
    }

    // ---- O^T += V^T x P : 8 d-blocks x 4 K-steps of WMMA 16x16x4 ----
#pragma unroll
    for (int dblk = 0; dblk < 8; dblk++) {
      int d = dblk * 16 + lo;
      v8f o = acc[dblk];
#pragma unroll
      for (int t = 0; t < 4; t++) {
        v2f av; av.x = vr0[t][d]; av.y = vr1[t][d];
        o = __builtin_amdgcn_wmma_f32_16x16x4_f32(false, av, false, pb[t], (short)0, o, false, false);
      }
      acc[dblk] = o;
    }
  }

  // ---- cross-wave flash combine via LDS ----
  if (lo < G_SZ) {
    if (hi == 0) { Rm[w][lo] = m_run; Rl[w][lo] = l_run; }
#pragma unroll
    for (int dblk = 0; dblk < 8; dblk++)
#pragma unroll
      for (int j = 0; j < 8; j++)
        Racc[w][lo][dblk * 16 + j + 8 * hi] = acc[dblk][j];
  }
  __syncthreads();

  for (int i = tid; i < G_SZ * D_SZ; i += THREADS) {
    int h = i >> 7;
    int d = i & 127;
    float mmax = NEG_BIG;
#pragma unroll
    for (int ww = 0; ww < NW; ww++) mmax = fmaxf(mmax, Rm[ww][h]);
    float num = 0.f, den = 0.f;
#pragma unroll
    for (int ww = 0; ww < NW; ww++) {
      float e = __expf(Rm[ww][h] - mmax);
      num += e * Racc[ww][h][d];
      den += e * Rl[ww][h];
    }
    if (nsplits == 1) {
      out[((size_t)b * H_SZ + kvh * G_SZ + h) * D_SZ + d] = num / den;
    } else {
      float* p = partial + ((size_t)bk * nsplits + split) * PARTSZ;
      if (d == 0) { p[h] = mmax; p[G_SZ + h] = den; }
      p[8 + h * D_SZ + d] = num;
    }
  }
}

__global__ __launch_bounds__(256) void pa_combine_kernel(
    const float* __restrict__ partial, float* __restrict__ out, int nsplits)
{
  int bk = blockIdx.x;
  int b = bk / KVH_SZ, kvh = bk % KVH_SZ;
  const float* base = partial + (size_t)bk * nsplits * PARTSZ;
  for (int i = threadIdx.x; i < G_SZ * D_SZ; i += 256) {
    int h = i >> 7, d = i & 127;
    float mmax = NEG_BIG;
    for (int s = 0; s < nsplits; s++)
      mmax = fmaxf(mmax, base[s * PARTSZ + h]);
    float num = 0.f, den = 0.f;
    for (int s = 0; s < nsplits; s++) {
      const float* p = base + s * PARTSZ;
      float e = __expf(p[h] - mmax);
      num += e * p[8 + h * D_SZ + d];
      den += e * p[G_SZ + h];
    }
    out[((size_t)b * H_SZ + kvh * G_SZ + h) * D_SZ + d] = num / den;
  }
}

extern "C" void kernel_launch(void* const* d_in, const int* in_sizes, int n_in,
                              void* d_out, int out_size, void* d_ws, size_t ws_size,
                              hipStream_t stream) {
  (void)in_sizes; (void)n_in; (void)out_size;
  const float* q     = (const float*)d_in[0];
  const float* kc    = (const float*)d_in[1];
  const float* vc    = (const float*)d_in[2];
  const int*   slots = (const int*)d_in[3];
  const int*   ctx   = (const int*)d_in[4];
  float* out = (float*)d_out;
  float* ws  = (float*)d_ws;

  int nsplits = 4;
  size_t need = (size_t)B_SZ * KVH_SZ * nsplits * PARTSZ * sizeof(float);
  if (ws_size < need) nsplits = 1;   // no workspace path: fully combined in-WG

  dim3 grid(B_SZ * KVH_SZ * nsplits);
  pa_main_kernel<<<grid, THREADS, 0, stream>>>(q, kc, vc, slots, ctx, out, ws, nsplits);
  if (nsplits > 1) {
    pa_combine_kernel<<<dim3(B_SZ * KVH_SZ), 256, 0, stream>>>(ws, out, nsplits);
  }
}